// NCEAverage_41755672052267
// MI455X (gfx1250) — compile-verified
//
#include <hip/hip_runtime.h>

// Problem constants (match reference)
#define BB    256        // batch
#define K1    4097       // K negatives + 1 positive
#define DD    128        // feature dim
#define NN    1000000    // memory bank rows
#define INV_T (1.0f / 0.07f)
#define MOM   0.5f

typedef float v2f __attribute__((ext_vector_type(2)));
typedef float v8f __attribute__((ext_vector_type(8)));

// ---------------------------------------------------------------------------
// Kernel 1: scores.  One workgroup (8 wave32) per batch element b.
// Each wave processes 16-row k-tiles with V_WMMA_F32_16X16X4_F32:
//   A (16x4)  = gathered memory rows, K-chunked by 4
//   B (4x16)  = columns 0..2 hold ab/l/ss vectors (rest zero-padded in LDS)
//   C (16x16) = f32 scores, columns 0..2 live
// ---------------------------------------------------------------------------
__global__ __launch_bounds__(256)
void nce_scores_kernel(const float* __restrict__ ab,
                       const float* __restrict__ l,
                       const float* __restrict__ ss,
                       const float* __restrict__ mem,
                       const int*  __restrict__ idx,
                       float*      __restrict__ out)
{
    const int b = blockIdx.x;

    // B-matrix staging: 16 columns x 128 K-values; cols >= 3 are zero so the
    // WMMA B-fragment build is branchless and EXEC stays all-ones.
    __shared__ float xs[16][DD];   // 8 KB
    for (int i = threadIdx.x; i < 16 * DD; i += blockDim.x) {
        const int v = i >> 7;      // column
        const int d = i & (DD - 1);
        float val = 0.0f;
        if (v == 0)      val = ab[(size_t)b * DD + d];
        else if (v == 1) val = l [(size_t)b * DD + d];
        else if (v == 2) val = ss[(size_t)b * DD + d];
        xs[v][d] = val;
    }
    __syncthreads();

    const int lane = threadIdx.x & 31;
    const int wave = threadIdx.x >> 5;
    const int half = lane >> 4;    // 0: K={0,1}, 1: K={2,3} of each chunk
    const int sub  = lane & 15;    // A: row-in-tile; B/C: column

    const int num_tiles = (K1 + 15) / 16;   // 257

    for (int t = wave; t < num_tiles; t += 8) {
        // This lane's gathered A-row (clamped; duplicates masked at store)
        int k = t * 16 + sub;
        if (k > K1 - 1) k = K1 - 1;
        const int   row  = idx[(size_t)b * K1 + k];
        const float* __restrict__ wrow = mem + (size_t)row * DD;

        v8f c = {0.f, 0.f, 0.f, 0.f, 0.f, 0.f, 0.f, 0.f};

        #pragma unroll
        for (int kk = 0; kk < DD; kk += 4) {
            // A fragment: lanes 0-15 load cols kk..kk+1, lanes 16-31 kk+2..kk+3
            v2f a = *(const v2f*)(wrow + kk + 2 * half);
            // B fragment: column 'sub'; VGPR0 holds K={0,1}, VGPR1 K={2,3}
            v2f bvec;
            bvec.x = xs[sub][kk + half];
            bvec.y = xs[sub][kk + 2 + half];
            c = __builtin_amdgcn_wmma_f32_16x16x4_f32(
                    /*neg_a=*/false, a, /*neg_b=*/false, bvec,
                    /*c_mod=*/(short)0, c, /*reuse_a=*/false, /*reuse_b=*/false);
        }

        // C layout: lane column n = sub; VGPR r holds row M = r + 8*half.
        if (sub < 3) {
            #pragma unroll
            for (int r = 0; r < 8; ++r) {
                const int m  = r + 8 * half;
                const int kq = t * 16 + m;
                if (kq < K1) {
                    out[(size_t)sub * BB * K1 + (size_t)b * K1 + kq] =
                        c[r] * INV_T;
                }
            }
        }
    }
}

// ---------------------------------------------------------------------------
// Kernel 2: stream-copy the 512 MB bank into the output (grid-stride float4).
// ---------------------------------------------------------------------------
__global__ __launch_bounds__(256)
void bank_copy_kernel(const float4* __restrict__ src,
                      float4*       __restrict__ dst,
                      long long n4)
{
    long long i      = (long long)blockIdx.x * blockDim.x + threadIdx.x;
    long long stride = (long long)gridDim.x * blockDim.x;
    for (; i < n4; i += stride) dst[i] = src[i];
}

// ---------------------------------------------------------------------------
// Kernel 3: momentum update of the 256 selected rows, one wave32 per row.
// Reads the OLD bank (d_in, never mutated), writes the copied output bank.
// ---------------------------------------------------------------------------
__global__ __launch_bounds__(32)
void bank_update_kernel(const float* __restrict__ ab,
                        const float* __restrict__ mem,
                        const int*  __restrict__ y,
                        float*      __restrict__ newmem)
{
    const int b = blockIdx.x;
    const int t = threadIdx.x;            // 0..31, each owns 4 floats
    const size_t row = (size_t)y[b] * DD;

    float4 m = *(const float4*)(mem + row + 4 * t);
    float4 a = *(const float4*)(ab + (size_t)b * DD + 4 * t);

    float4 p;
    p.x = m.x * MOM + a.x * (1.0f - MOM);
    p.y = m.y * MOM + a.y * (1.0f - MOM);
    p.z = m.z * MOM + a.z * (1.0f - MOM);
    p.w = m.w * MOM + a.w * (1.0f - MOM);

    float s = p.x * p.x + p.y * p.y + p.z * p.z + p.w * p.w;
    // wave32 butterfly reduction (warpSize == 32 on gfx1250)
    for (int off = 16; off > 0; off >>= 1)
        s += __shfl_xor(s, off, 32);

    const float inv = 1.0f / sqrtf(s);
    p.x *= inv; p.y *= inv; p.z *= inv; p.w *= inv;

    *(float4*)(newmem + row + 4 * t) = p;
}

// ---------------------------------------------------------------------------
extern "C" void kernel_launch(void* const* d_in, const int* in_sizes, int n_in,
                              void* d_out, int out_size, void* d_ws, size_t ws_size,
                              hipStream_t stream)
{
    const float* ab  = (const float*)d_in[0];   // [B, D]
    const float* l   = (const float*)d_in[1];   // [B, D]
    const float* ss  = (const float*)d_in[2];   // [B, D]
    const float* mem = (const float*)d_in[3];   // [N, D]
    const int*   idx = (const int*)  d_in[4];   // [B, K1]
    const int*   y   = (const int*)  d_in[5];   // [B]

    float* out    = (float*)d_out;                       // [3, B, K1, 1]
    float* newmem = out + (size_t)3 * BB * K1;           // [N, D]

    // 1) scores via f32 WMMA
    nce_scores_kernel<<<BB, 256, 0, stream>>>(ab, l, ss, mem, idx, out);

    // 2) copy bank (HBM-streaming), then 3) scatter the updated rows
    const long long n4 = (long long)NN * DD / 4;         // 32M float4
    bank_copy_kernel<<<8192, 256, 0, stream>>>(
        (const float4*)mem, (float4*)newmem, n4);
    bank_update_kernel<<<BB, 32, 0, stream>>>(ab, mem, y, newmem);
}